// IndRNNCell_64561948393994
// MI455X (gfx1250) — compile-verified
//
#include <hip/hip_runtime.h>
#include <hip/hip_bf16.h>

typedef __attribute__((ext_vector_type(2))) float v2f;
typedef __attribute__((ext_vector_type(8))) float v8f;

// ---------------------------------------------------------------------------
// Kernel 1: proj = x @ W + bias, written straight into d_out.
//   x:    [32768, 256]  (B*T rows, D cols)
//   W:    [256, 512]
//   bias: [512]
//   out:  [32768, 512]
// Block: 256 threads = 8 waves as 4(m) x 2(n); block tile M=64, N=64, K=256.
// Each wave computes a 16x32 strip via two 16x16 accumulators using
// V_WMMA_F32_16X16X4_F32 (full fp32 — precision-exact, and compute is far
// below the memory-bound roofline so fp32 WMMA is free).
//
// W is staged into LDS k-pair interleaved: Wp[kp][n] = (W[2kp][n], W[2kp+1][n])
// so each B fragment is ONE ds_load_b64 straight into an adjacent even VGPR
// pair (WMMA operand requirement) — no packing movs.
// ---------------------------------------------------------------------------
#define LDA  260   // 256 + 4 pad: A-frag ds_load_b64 conflict-free (bank 4m)
#define LDWP 80    // float2/row: 160 dwords ≡ 32 banks (mod 64) → half-waves
                   // hit disjoint 32-bank windows on B-frag ds_load_b64

__global__ __launch_bounds__(256) void indrnn_gemm_kernel(
    const float* __restrict__ x, const float* __restrict__ W,
    const float* __restrict__ bias, float* __restrict__ out)
{
    __shared__ float  As[64 * LDA];    // 66,560 B
    __shared__ float2 Wp[128 * LDWP];  // 81,920 B  (total 148,480 B < 320 KB)

    const int tid  = threadIdx.x;
    const int lane = tid & 31;
    const int wave = tid >> 5;       // 0..7
    const int wm   = wave >> 1;      // 0..3 : 16-row strip
    const int wn   = wave & 1;       // 0..1 : 32-col half
    const int m0   = blockIdx.x * 64;
    const int n0   = blockIdx.y * 64;

    // ---- Stage A tile (64 x 256) with coalesced float4 loads ----
    {
        const float4* xv = (const float4*)(x + (size_t)m0 * 256);
        #pragma unroll
        for (int i = 0; i < 16; ++i) {
            int idx = tid + i * 256;           // 0..4095 float4s
            int row = idx >> 6;                // /64 float4 per row
            int c4  = idx & 63;
            float4 v = xv[idx];
            *(float4*)&As[row * LDA + c4 * 4] = v;
        }
    }
    // ---- Stage W tile (256 x 64) k-pair interleaved ----
    {
        const float* Wg = W + n0;
        #pragma unroll
        for (int i = 0; i < 8; ++i) {
            int idx = tid + i * 256;           // 0..2047
            int kp  = idx >> 4;                // 0..127 : pair of rows (2kp,2kp+1)
            int c   = (idx & 15) * 4;          // col group
            float4 r0 = *(const float4*)(Wg + (size_t)(2 * kp)     * 512 + c);
            float4 r1 = *(const float4*)(Wg + (size_t)(2 * kp + 1) * 512 + c);
            float4 q0 = make_float4(r0.x, r1.x, r0.y, r1.y);
            float4 q1 = make_float4(r0.z, r1.z, r0.w, r1.w);
            float* dst = (float*)&Wp[kp * LDWP + c];
            *(float4*)(dst)     = q0;
            *(float4*)(dst + 4) = q1;
        }
    }
    __syncthreads();

    // Fragment layout (ISA 7.12.2, 32-bit A 16x4 / B 4x16):
    //   lane l: M (or N) = l&15 ; VGPR pair holds K = {kb, kb+1}, kb = 2*(l>>4)
    const int half = lane >> 4;
    const int ml   = lane & 15;
    const int kb   = half * 2;

    // C init = bias broadcast (C layout: all 8 regs of a lane share N = l&15)
    v8f acc[2];
    #pragma unroll
    for (int nt = 0; nt < 2; ++nt) {
        float bv = bias[n0 + wn * 32 + nt * 16 + ml];
        #pragma unroll
        for (int r = 0; r < 8; ++r) acc[nt][r] = bv;
    }

    const float*  arow  = &As[(wm * 16 + ml) * LDA + kb];
    const float2* wbase = &Wp[(size_t)half * LDWP + wn * 32 + ml];
    #pragma unroll 4
    for (int kk = 0; kk < 64; ++kk) {
        v2f a = *(const v2f*)(arow + kk * 4);                    // ds_load_b64
        #pragma unroll
        for (int nt = 0; nt < 2; ++nt) {
            // K rows {4kk+kb, 4kk+kb+1} = pair index 2kk+half  -> one b64 load
            v2f bf = *(const v2f*)(wbase + (size_t)(2 * kk) * LDWP + nt * 16);
            acc[nt] = __builtin_amdgcn_wmma_f32_16x16x4_f32(
                false, a, false, bf, (short)0, acc[nt], false, false);
        }
    }

    // Epilogue: C/D layout -> out. Lanes 0..15 write 64 contiguous bytes/row.
    const int row0 = m0 + wm * 16 + half * 8;
    #pragma unroll
    for (int nt = 0; nt < 2; ++nt) {
        const int col = n0 + wn * 32 + nt * 16 + ml;
        #pragma unroll
        for (int r = 0; r < 8; ++r) {
            out[(size_t)(row0 + r) * 512 + col] = acc[nt][r];
        }
    }
}

// ---------------------------------------------------------------------------
// Kernel 2: in-place IndRNN scan over T on d_out.
//   d_out currently holds proj[b,t,u]; each element is read once and then
//   overwritten by h[b,t,u] at the same address (read-before-write).
// One thread per (b, 4 columns): 64*128 = 8192 threads. proj loads are
// h-independent, so prefetch row t+1 while the dependent relu chain retires.
// ---------------------------------------------------------------------------
__global__ __launch_bounds__(256) void indrnn_scan_kernel(
    float* __restrict__ hb, const float* __restrict__ h0,
    const float* __restrict__ u)
{
    const int tid = blockIdx.x * 256 + threadIdx.x;    // 0..8191
    const int b   = tid >> 7;                          // 0..63
    const int col = (tid & 127) * 4;                   // 0,4,...,508

    float4 uc = *(const float4*)(u + col);
    uc.x = fminf(fmaxf(uc.x, 0.0f), 1.0f);
    uc.y = fminf(fmaxf(uc.y, 0.0f), 1.0f);
    uc.z = fminf(fmaxf(uc.z, 0.0f), 1.0f);
    uc.w = fminf(fmaxf(uc.w, 0.0f), 1.0f);

    float4 h = *(const float4*)(h0 + (size_t)b * 512 + col);

    float* base = hb + (size_t)b * 512 * 512 + col;
    float4 p = *(float4*)base;                         // t = 0

    #pragma unroll 8
    for (int t = 0; t < 511; ++t) {
        float4 pn = *(float4*)(base + (size_t)(t + 1) * 512);  // prefetch
        h.x = fmaxf(fmaf(h.x, uc.x, p.x), 0.0f);
        h.y = fmaxf(fmaf(h.y, uc.y, p.y), 0.0f);
        h.z = fmaxf(fmaf(h.z, uc.z, p.z), 0.0f);
        h.w = fmaxf(fmaf(h.w, uc.w, p.w), 0.0f);
        *(float4*)(base + (size_t)t * 512) = h;
        p = pn;
    }
    // final step t = 511
    h.x = fmaxf(fmaf(h.x, uc.x, p.x), 0.0f);
    h.y = fmaxf(fmaf(h.y, uc.y, p.y), 0.0f);
    h.z = fmaxf(fmaf(h.z, uc.z, p.z), 0.0f);
    h.w = fmaxf(fmaf(h.w, uc.w, p.w), 0.0f);
    *(float4*)(base + (size_t)511 * 512) = h;
}

// ---------------------------------------------------------------------------
extern "C" void kernel_launch(void* const* d_in, const int* in_sizes, int n_in,
                              void* d_out, int out_size, void* d_ws, size_t ws_size,
                              hipStream_t stream) {
    (void)in_sizes; (void)n_in; (void)out_size; (void)d_ws; (void)ws_size;
    const float* x    = (const float*)d_in[0];   // [64,512,256]
    const float* h0   = (const float*)d_in[1];   // [64,512]
    const float* W    = (const float*)d_in[2];   // [256,512]
    const float* u    = (const float*)d_in[3];   // [512]
    const float* bias = (const float*)d_in[4];   // [512]
    float* out = (float*)d_out;                  // [64,512,512]

    dim3 g1(32768 / 64, 512 / 64);               // 512 x 8 blocks
    indrnn_gemm_kernel<<<g1, 256, 0, stream>>>(x, W, bias, out);
    indrnn_scan_kernel<<<32, 256, 0, stream>>>(out, h0, u);
}